// GLIF_69922067579062
// MI455X (gfx1250) — compile-verified
//
#include <hip/hip_runtime.h>
#include <math.h>

// GLIF: T=4 elementwise recurrence over (16, 2048, 1024) fp32.
// Memory-bound streaming kernel: 128MiB in + 128MiB out @ 23.3TB/s => ~11.5us floor.
// gfx1250 path used: double-buffered GLOBAL_LOAD_ASYNC_TO_LDS_B128 + S_WAIT_ASYNCCNT.

#define CCH     2048
#define LEN     1024
#define TSTEPS  4
#define ROWS    4      // (b,c) rows per block
#define THREADS 256    // 8 wave32s; 256 lanes x float4 == one full row of L=1024

#if defined(__AMDGCN__) && __has_builtin(__builtin_amdgcn_global_load_async_to_lds_b128) && __has_builtin(__builtin_amdgcn_s_wait_asynccnt)
#define USE_ASYNC_LDS 1
#else
#define USE_ASYNC_LDS 0
#endif

// Matches the parameter type hipcc reported for the async builtin:
// '__attribute__((__vector_size__(4 * sizeof(int)))) int *'
typedef int v4i __attribute__((vector_size(16)));

__device__ __forceinline__ float sigf(float v) {
    return 1.0f / (1.0f + __expf(-v));
}

// 16B per-lane async DMA global -> LDS (ASYNCcnt), falls back to a plain copy.
__device__ __forceinline__ void async_copy16(const float* gsrc, float* ldst) {
#if USE_ASYNC_LDS
    __builtin_amdgcn_global_load_async_to_lds_b128(
        (v4i*)gsrc, (v4i*)ldst, /*offset=*/0, /*cpol=*/0);
#else
    *(float4*)ldst = *(const float4*)gsrc;
#endif
}

__device__ __forceinline__ void wait_async_le(int n) {
#if USE_ASYNC_LDS
    if (n == 0) __builtin_amdgcn_s_wait_asynccnt(0);
    else        __builtin_amdgcn_s_wait_asynccnt(1);
#else
    (void)n;
#endif
}

__global__ __launch_bounds__(THREADS) void glif_scan_kernel(
    const float* __restrict__ x,
    const float* __restrict__ p_alpha, const float* __restrict__ p_beta,
    const float* __restrict__ p_gamma, const float* __restrict__ p_tau,
    const float* __restrict__ p_vth,   const float* __restrict__ p_leak,
    const float* __restrict__ p_revth, const float* __restrict__ p_cond,
    float* __restrict__ out)
{
    __shared__ float buf[2][THREADS * 4];   // 2 x 4KiB double buffer (one row each)

    const int tid     = threadIdx.x;
    const int rowBase = blockIdx.x * ROWS;          // global row = b*CCH + c
    const int lane4   = tid * 4;                    // float offset within row

    // Prologue: stage row 0 into buffer 0 (each lane DMAs exactly its own 16B).
    async_copy16(x + (size_t)rowBase * LEN + lane4, &buf[0][lane4]);

#pragma unroll
    for (int r = 0; r < ROWS; ++r) {
        const int row = rowBase + r;
        const int c   = row & (CCH - 1);            // channel id (uniform per row)

        // Prefetch next row into the other buffer while this row computes.
        if (r + 1 < ROWS)
            async_copy16(x + (size_t)(row + 1) * LEN + lane4, &buf[(r + 1) & 1][lane4]);

        // Row r's DMA is the oldest outstanding async op.
        wait_async_le((r + 1 < ROWS) ? 1 : 0);

        const float4 xv = ((const float4*)buf[r & 1])[tid];
        const float xs[4] = { xv.x, xv.y, xv.z, xv.w };

        // ---- per-channel (block-uniform) constants ----
        const bool alb = p_alpha[c] > 0.0f;         // arch_act(sigmoid(v)) == (v > 0)
        const bool beb = p_beta [c] > 0.0f;
        const bool gab = p_gamma[c] > 0.0f;
        const float tau_s = sigf(p_tau[c]);
        const float vth_s = sigf(p_vth[c]);
        const float A  = alb ? tau_s : 1.0f;        // 1 - al*(1-tau)
        const float Lk = alb ? 0.0f  : sigf(p_leak[c]);   // (1-al)*leak
        const float R  = gab ? 0.0f  : sigf(p_revth[c]);  // (1-ga)*reVth
        float W[TSTEPS];
#pragma unroll
        for (int t = 0; t < TSTEPS; ++t)
            W[t] = beb ? sigf(p_cond[t * CCH + c]) : 1.0f;  // 1 - be*(1-c_t)

        // ---- T=4 register scan; gab is uniform -> scalar branch ----
        float os[4];
        if (gab) {
            // u = A*u*(1-o) + (W_t*x - Lk);  o = u > Vth
#pragma unroll
            for (int j = 0; j < 4; ++j) {
                float u = 0.0f, o = 0.0f;
#pragma unroll
                for (int t = 0; t < TSTEPS; ++t) {
                    const float xw = fmaf(W[t], xs[j], -Lk);
                    u = fmaf(A * u, 1.0f - o, xw);
                    o = (u > vth_s) ? 1.0f : 0.0f;
                }
                os[j] = o;
            }
        } else {
            // u = A*u + (W_t*x - Lk) - R*o;  o = u > Vth
#pragma unroll
            for (int j = 0; j < 4; ++j) {
                float u = 0.0f, o = 0.0f;
#pragma unroll
                for (int t = 0; t < TSTEPS; ++t) {
                    const float xw = fmaf(W[t], xs[j], -Lk);
                    u = fmaf(A, u, fmaf(-R, o, xw));
                    o = (u > vth_s) ? 1.0f : 0.0f;
                }
                os[j] = o;
            }
        }

        ((float4*)out)[(size_t)row * (LEN / 4) + tid] =
            make_float4(os[0], os[1], os[2], os[3]);
    }
}

extern "C" void kernel_launch(void* const* d_in, const int* in_sizes, int n_in,
                              void* d_out, int out_size, void* d_ws, size_t ws_size,
                              hipStream_t stream) {
    (void)in_sizes; (void)n_in; (void)out_size; (void)d_ws; (void)ws_size;
    const float* x       = (const float*)d_in[0];
    const float* alpha   = (const float*)d_in[1];
    const float* beta    = (const float*)d_in[2];
    const float* gamma_  = (const float*)d_in[3];
    const float* tau     = (const float*)d_in[4];
    const float* vth     = (const float*)d_in[5];
    const float* leak    = (const float*)d_in[6];
    const float* revth   = (const float*)d_in[7];
    const float* conduct = (const float*)d_in[8];
    float* out = (float*)d_out;

    const int totalRows = 16 * CCH;                 // B * C = 32768 rows of L=1024
    dim3 grid(totalRows / ROWS);                    // 8192 blocks
    dim3 block(THREADS);                            // 8 wave32s
    glif_scan_kernel<<<grid, block, 0, stream>>>(
        x, alpha, beta, gamma_, tau, vth, leak, revth, conduct, out);
}